// RPN_65695819759989
// MI455X (gfx1250) — compile-verified
//
#include <hip/hip_runtime.h>
#include <math.h>

// ---------------- problem constants ----------------
#define B_IMGS   4
#define A_NUM    3
#define H0       100
#define W0       100
#define H1       50
#define W1       50
#define N0       30000          // 100*100*3
#define N1       7500           // 50*50*3
#define N_TOT    37500
#define K_LVL    2000
#define K_TOT    4000
#define POST_TOP 1000
#define NMS_TH   0.7f
#define MIN_SZ   1e-3f
#define BBOX_CLIP 4.1351665567423560f   // log(1000/16)
#define NW       125                    // 4000/32 mask words per row
#define SORT_N   4096

typedef float v8f __attribute__((ext_vector_type(8)));
typedef float v2f __attribute__((ext_vector_type(2)));

__device__ __forceinline__ unsigned fkey(float f) {
    // monotonic float->uint transform (ascending)
    unsigned u = __float_as_uint(f);
    return u ^ ((u >> 31) ? 0xFFFFFFFFu : 0x80000000u);
}

// ---------------- kernel 0: init scratch ----------------
__global__ void k_init(unsigned* maxc, int* top_index) {
    int i = blockIdx.x * blockDim.x + threadIdx.x;
    if (i < B_IMGS) maxc[i] = 0u;
    if (i < B_IMGS * K_TOT) top_index[i] = 0;
}

// ---------------- kernel 1: format + decode ----------------
// scores_all[b*N + n] in (h,w,anchor) order; proposals decoded for LAST image only
// (reference quirk: gathers scores/boxes from proposals[-1]).
__global__ void k_decode(const float* __restrict__ cls0, const float* __restrict__ reg0,
                         const float* __restrict__ cls1, const float* __restrict__ reg1,
                         const float* __restrict__ an0,  const float* __restrict__ an1,
                         float* __restrict__ scores_all, float* __restrict__ prop_last) {
    int n = blockIdx.x * blockDim.x + threadIdx.x;
    if (n >= N_TOT) return;
    float ax1, ay1, ax2, ay2, r0v, r1v, r2v, r3v;
    if (n < N0) {
        int local = n;
        int a   = local % A_NUM;
        int pos = local / A_NUM;
        int h = pos / W0, w = pos % W0;
        ax1 = an0[local * 4 + 0]; ay1 = an0[local * 4 + 1];
        ax2 = an0[local * 4 + 2]; ay2 = an0[local * 4 + 3];
        #pragma unroll
        for (int b = 0; b < B_IMGS; ++b)
            scores_all[b * N_TOT + n] = cls0[((b * A_NUM + a) * H0 + h) * W0 + w];
        const int bb = B_IMGS - 1;
        r0v = reg0[((bb * A_NUM * 4 + a * 4 + 0) * H0 + h) * W0 + w];
        r1v = reg0[((bb * A_NUM * 4 + a * 4 + 1) * H0 + h) * W0 + w];
        r2v = reg0[((bb * A_NUM * 4 + a * 4 + 2) * H0 + h) * W0 + w];
        r3v = reg0[((bb * A_NUM * 4 + a * 4 + 3) * H0 + h) * W0 + w];
    } else {
        int local = n - N0;
        int a   = local % A_NUM;
        int pos = local / A_NUM;
        int h = pos / W1, w = pos % W1;
        ax1 = an1[local * 4 + 0]; ay1 = an1[local * 4 + 1];
        ax2 = an1[local * 4 + 2]; ay2 = an1[local * 4 + 3];
        #pragma unroll
        for (int b = 0; b < B_IMGS; ++b)
            scores_all[b * N_TOT + n] = cls1[((b * A_NUM + a) * H1 + h) * W1 + w];
        const int bb = B_IMGS - 1;
        r0v = reg1[((bb * A_NUM * 4 + a * 4 + 0) * H1 + h) * W1 + w];
        r1v = reg1[((bb * A_NUM * 4 + a * 4 + 1) * H1 + h) * W1 + w];
        r2v = reg1[((bb * A_NUM * 4 + a * 4 + 2) * H1 + h) * W1 + w];
        r3v = reg1[((bb * A_NUM * 4 + a * 4 + 3) * H1 + h) * W1 + w];
    }
    float wa = ax2 - ax1, ha = ay2 - ay1;
    float cx = ax1 + 0.5f * wa, cy = ay1 + 0.5f * ha;
    float dw = fminf(r2v, BBOX_CLIP), dh = fminf(r3v, BBOX_CLIP);
    float px = cx + r0v * wa, py = cy + r1v * ha;
    float pw = expf(dw) * wa, ph = expf(dh) * ha;
    prop_last[n * 4 + 0] = px - 0.5f * pw;
    prop_last[n * 4 + 1] = py - 0.5f * ph;
    prop_last[n * 4 + 2] = px + 0.5f * pw;
    prop_last[n * 4 + 3] = py + 0.5f * ph;
}

// ---------------- kernel 2: per-(image,level) top-K radix select ----------------
__global__ __launch_bounds__(256) void k_topk(const float* __restrict__ scores_all,
                                              int* __restrict__ top_index) {
    int blk = blockIdx.x;                 // 0..7
    int b = blk >> 1, lvl = blk & 1;
    int count = lvl ? N1 : N0;
    int base  = b * N_TOT + (lvl ? N0 : 0);
    int gbase = lvl ? N0 : 0;
    int tid = threadIdx.x;

    __shared__ unsigned hist[256];
    __shared__ unsigned sh_pref, sh_want, sh_cum;
    __shared__ unsigned ctrG, ctrE;

    unsigned pref = 0, prefmask = 0, want = K_LVL, totalAbove = 0;
    for (int round = 0; round < 4; ++round) {
        int shift = 24 - 8 * round;
        if (tid < 256) hist[tid] = 0;
        __syncthreads();
        for (int i = tid; i < count; i += 256) {
            unsigned key = fkey(scores_all[base + i]);
            if ((key & prefmask) == pref)
                atomicAdd(&hist[(key >> shift) & 255], 1u);
        }
        __syncthreads();
        if (tid == 0) {
            unsigned cum = 0; int d = 255;
            for (; d > 0; --d) {
                unsigned c = hist[d];
                if (cum + c >= want) break;
                cum += c;
            }
            sh_pref = pref | ((unsigned)d << shift);
            sh_want = want - cum;
            sh_cum  = cum;
        }
        __syncthreads();
        pref = sh_pref;
        want = sh_want;
        totalAbove += sh_cum;
        prefmask |= (0xFFu << shift);
        __syncthreads();
    }
    unsigned T = pref, cg = totalAbove, wantEq = want;
    if (tid == 0) { ctrG = 0u; ctrE = 0u; }
    __syncthreads();
    int* dst = top_index + b * K_TOT + lvl * K_LVL;
    for (int i = tid; i < count; i += 256) {
        unsigned key = fkey(scores_all[base + i]);
        if (key > T) {
            unsigned p = atomicAdd(&ctrG, 1u);
            dst[p] = gbase + i;
        } else if (key == T) {
            unsigned p = atomicAdd(&ctrE, 1u);
            if (p < wantEq) dst[cg + p] = gbase + i;
        }
    }
}

// ---------------- kernel 3: gather (from last image) + clip + max-coord ----------------
__global__ void k_gather(const int* __restrict__ top_index, const float* __restrict__ scores_all,
                         const float* __restrict__ prop_last,
                         const int* __restrict__ ih_p, const int* __restrict__ iw_p,
                         float* __restrict__ box_g, float* __restrict__ score_g,
                         int* __restrict__ lvl_g, int* __restrict__ valid_g,
                         unsigned* __restrict__ maxc) {
    int b = blockIdx.y;
    int i = blockIdx.x * blockDim.x + threadIdx.x;
    if (i >= K_TOT) return;
    float img_h = (float)ih_p[0], img_w = (float)iw_p[0];
    int n = top_index[b * K_TOT + i];
    n = (n < 0) ? 0 : ((n >= N_TOT) ? N_TOT - 1 : n);
    float s  = scores_all[(B_IMGS - 1) * N_TOT + n];
    float x1 = fminf(fmaxf(prop_last[n * 4 + 0], 0.f), img_w);
    float y1 = fminf(fmaxf(prop_last[n * 4 + 1], 0.f), img_h);
    float x2 = fminf(fmaxf(prop_last[n * 4 + 2], 0.f), img_w);
    float y2 = fminf(fmaxf(prop_last[n * 4 + 3], 0.f), img_h);
    int valid = (x2 - x1 >= MIN_SZ) && (y2 - y1 >= MIN_SZ);
    int o = b * K_TOT + i;
    box_g[o * 4 + 0] = x1; box_g[o * 4 + 1] = y1;
    box_g[o * 4 + 2] = x2; box_g[o * 4 + 3] = y2;
    score_g[o] = valid ? s : -INFINITY;
    lvl_g[o]   = (n >= N0) ? 1 : 0;
    valid_g[o] = valid;
    float m4 = valid ? fmaxf(fmaxf(x1, y1), fmaxf(x2, y2)) : 0.f;  // coords >= 0
    atomicMax(&maxc[b], __float_as_uint(m4));
}

// ---------------- kernel 4: per-image bitonic sort in LDS (desc score, asc idx) ----------------
__global__ __launch_bounds__(512) void k_sort(const float* __restrict__ box_g,
                                              const float* __restrict__ score_g,
                                              const int* __restrict__ lvl_g,
                                              const int* __restrict__ valid_g,
                                              const unsigned* __restrict__ maxc,
                                              float* __restrict__ box_s, float* __restrict__ score_s,
                                              int* __restrict__ valid_s, float* __restrict__ nb_s) {
    int b = blockIdx.x;
    int tid = threadIdx.x;
    const int T = 512;
    __shared__ float sk[SORT_N];
    __shared__ int   si[SORT_N];
    for (int i = tid; i < SORT_N; i += T) {
        sk[i] = (i < K_TOT) ? score_g[b * K_TOT + i] : -INFINITY;
        si[i] = i;
    }
    __syncthreads();
    for (int k = 2; k <= SORT_N; k <<= 1) {
        for (int j = k >> 1; j > 0; j >>= 1) {
            for (int i = tid; i < SORT_N; i += T) {
                int ixj = i ^ j;
                if (ixj > i) {
                    float a = sk[i], c = sk[ixj];
                    int ia = si[i], ic = si[ixj];
                    bool aFirst = (a > c) || (a == c && ia < ic);  // desired: desc, idx asc
                    bool descBlock = ((i & k) == 0);
                    bool doSwap = descBlock ? !aFirst : aFirst;
                    if (doSwap) { sk[i] = c; sk[ixj] = a; si[i] = ic; si[ixj] = ia; }
                }
            }
            __syncthreads();
        }
    }
    float mx = __uint_as_float(maxc[b]);
    float offmul = mx + 1.0f;
    for (int i = tid; i < K_TOT; i += T) {
        int o = si[i];
        int src = b * K_TOT + o;
        int dst = b * K_TOT + i;
        float off = (float)lvl_g[src] * offmul;
        float x1 = box_g[src * 4 + 0], y1 = box_g[src * 4 + 1];
        float x2 = box_g[src * 4 + 2], y2 = box_g[src * 4 + 3];
        box_s[dst * 4 + 0] = x1; box_s[dst * 4 + 1] = y1;
        box_s[dst * 4 + 2] = x2; box_s[dst * 4 + 3] = y2;
        nb_s[dst * 4 + 0] = x1 + off; nb_s[dst * 4 + 1] = y1 + off;
        nb_s[dst * 4 + 2] = x2 + off; nb_s[dst * 4 + 3] = y2 + off;
        score_s[dst] = sk[i];
        valid_s[dst] = valid_g[src];
    }
}

// ---------------- kernel 5: IoU suppression bitmask via WMMA + wave32 ballots ----------------
// Col boxes staged into LDS with gfx1250 async DMA (global_load_async_to_lds_b128,
// ASYNCcnt). Union term area[i]+area[j] produced on the matrix pipe as a rank-2
// outer sum with V_WMMA_F32_16X16X4_F32; IoU threshold as a divide-free fused
// multiply-compare; wave32 ballots assemble 32-bit suppression words.
__global__ __launch_bounds__(256) void k_mask(const float* __restrict__ nb_s,
                                              unsigned* __restrict__ mask) {
    int img  = blockIdx.z;
    int wcol = blockIdx.x;                 // 0..124 (32 cols each)
    int tid  = threadIdx.x;
    int wave = tid >> 5, lane = tid & 31;

    __shared__ float cb[32][4];
    __shared__ float ca[32];
    __shared__ float rbx[8][16][4];
    __shared__ float ra[8][16];

    const float* nbb = nb_s + (size_t)img * K_TOT * 4;

    // wave 0: async-DMA the 32 column boxes (16B/lane) straight into LDS.
    // LDS flat-aperture addresses carry the LDS byte offset in bits [31:0].
    if (tid < 32) {
        unsigned ldsoff = (unsigned)(uintptr_t)(&cb[0][0]) + (unsigned)tid * 16u;
        const float* gsrc = nbb + (size_t)(wcol * 32 + tid) * 4;
        asm volatile(
            "global_load_async_to_lds_b128 %0, %1, off\n\t"
            "s_wait_asynccnt 0"
            :
            : "v"(ldsoff), "v"(gsrc)
            : "memory");
    }

    int r0 = blockIdx.y * 128 + wave * 16;
    bool act = (r0 < K_TOT);               // wave-uniform
    if (act && lane < 16) {
        int r = r0 + lane;
        float x1 = nbb[r * 4 + 0], y1 = nbb[r * 4 + 1];
        float x2 = nbb[r * 4 + 2], y2 = nbb[r * 4 + 3];
        rbx[wave][lane][0] = x1; rbx[wave][lane][1] = y1;
        rbx[wave][lane][2] = x2; rbx[wave][lane][3] = y2;
        ra[wave][lane] = (x2 - x1) * (y2 - y1);
    }
    __syncthreads();
    if (tid < 32) ca[tid] = (cb[tid][2] - cb[tid][0]) * (cb[tid][3] - cb[tid][1]);
    __syncthreads();
    if (!act) return;

    // gfx1250 prefetch of next row-block's boxes (global_prefetch_b8)
    if (r0 + 128 < K_TOT) __builtin_prefetch(nbb + (size_t)(r0 + 128) * 4, 0, 0);

    // A: 16x4 f32 (lanes 0-15: K=0,1 in v0,v1; lanes 16-31: K=2,3 -> zero)
    v2f av, bv0, bv1;
    av.x  = (lane < 16) ? ra[wave][lane] : 0.f;   // K=0 column = row areas
    av.y  = (lane < 16) ? 1.f            : 0.f;   // K=1 column = ones
    bv0.x = (lane < 16) ? 1.f            : 0.f;   // K=0 row = ones
    bv0.y = (lane < 16) ? ca[lane]       : 0.f;   // K=1 row = col areas (cols 0-15)
    bv1.x = bv0.x;
    bv1.y = (lane < 16) ? ca[16 + lane]  : 0.f;   // cols 16-31
    v8f cz = {};
    v8f U0 = __builtin_amdgcn_wmma_f32_16x16x4_f32(false, av, false, bv0, (short)0, cz, false, false);
    v8f U1 = __builtin_amdgcn_wmma_f32_16x16x4_f32(false, av, false, bv1, (short)0, cz, false, false);

    int mrow_half = (lane < 16) ? 0 : 8;   // D layout: lanes 16-31 hold M = v+8
    int nl = lane & 15;
    unsigned* mrow = mask + ((size_t)img * K_TOT + r0) * NW + wcol;

    #pragma unroll
    for (int v = 0; v < 8; ++v) {
        int row = v + mrow_half;
        float rx1 = rbx[wave][row][0], ry1 = rbx[wave][row][1];
        float rx2 = rbx[wave][row][2], ry2 = rbx[wave][row][3];
        // half 0 (cols wcol*32 + 0..15)
        float iw0 = fmaxf(fminf(rx2, cb[nl][2]) - fmaxf(rx1, cb[nl][0]), 0.f);
        float ih0 = fmaxf(fminf(ry2, cb[nl][3]) - fmaxf(ry1, cb[nl][1]), 0.f);
        float inter0 = iw0 * ih0;
        float den0 = U0[v] - inter0;
        bool p0 = (den0 > 0.f) && (inter0 > NMS_TH * den0);   // == inter/den > 0.7, divide-free
        // half 1 (cols wcol*32 + 16..31)
        float iw1 = fmaxf(fminf(rx2, cb[16 + nl][2]) - fmaxf(rx1, cb[16 + nl][0]), 0.f);
        float ih1 = fmaxf(fminf(ry2, cb[16 + nl][3]) - fmaxf(ry1, cb[16 + nl][1]), 0.f);
        float inter1 = iw1 * ih1;
        float den1 = U1[v] - inter1;
        bool p1 = (den1 > 0.f) && (inter1 > NMS_TH * den1);

        unsigned m0 = (unsigned)__ballot(p0);   // bits[15:0]: row v; bits[31:16]: row v+8 (cols 0-15)
        unsigned m1 = (unsigned)__ballot(p1);   // same rows, cols 16-31
        if (lane == 0) {
            unsigned wlo = (m0 & 0xFFFFu) | ((m1 & 0xFFFFu) << 16);    // row v, full 32 cols
            unsigned whi = (m0 >> 16)     | (m1 & 0xFFFF0000u);        // row v+8, full 32 cols
            mrow[(size_t)v * NW]       = wlo;
            mrow[(size_t)(v + 8) * NW] = whi;
        }
    }
}

// ---------------- kernel 6: serial greedy scan (one wave32 per image) + compaction ----------------
// Validity preloaded into per-lane register bitmask words (wave32 ballots), so each
// serial iteration is register math + one shuffle; memory touched only on kept rows.
__global__ __launch_bounds__(32) void k_scan(const unsigned* __restrict__ mask,
                                             const float* __restrict__ box_s,
                                             const float* __restrict__ score_s,
                                             const int* __restrict__ valid_s,
                                             float* __restrict__ out_boxes,
                                             float* __restrict__ out_scores) {
    int img = blockIdx.x;
    int t = threadIdx.x;
    const unsigned* M = mask + (size_t)img * K_TOT * NW;
    const int* vs = valid_s + img * K_TOT;
    unsigned rem0 = 0, rem1 = 0, rem2 = 0, rem3 = 0;   // lane t owns words t, t+32, t+64, t+96
    unsigned kw0 = 0, kw1 = 0, kw2 = 0, kw3 = 0;
    unsigned vw0 = 0, vw1 = 0, vw2 = 0, vw3 = 0;

    // pack validity bits into the same word-ownership layout
    for (int w = 0; w < NW; ++w) {
        unsigned bw = (unsigned)__ballot(vs[w * 32 + t] != 0);
        if (t == (w & 31)) {
            int slot = w >> 5;
            if      (slot == 0) vw0 = bw;
            else if (slot == 1) vw1 = bw;
            else if (slot == 2) vw2 = bw;
            else                vw3 = bw;
        }
    }

    for (int i = 0; i < K_TOT; ++i) {
        int wi = i >> 5, bit = i & 31;
        int owner = wi & 31, slot = wi >> 5;
        unsigned kb = 0u;
        if (t == owner) {
            unsigned rsel = (slot == 0) ? rem0 : (slot == 1) ? rem1 : (slot == 2) ? rem2 : rem3;
            unsigned vsel = (slot == 0) ? vw0  : (slot == 1) ? vw1  : (slot == 2) ? vw2  : vw3;
            kb = (vsel >> bit) & ~(rsel >> bit) & 1u;
        }
        unsigned keep = (unsigned)__shfl((int)kb, owner, 32);
        if (keep) {
            if (t == owner) {
                unsigned b1 = 1u << bit;
                if      (slot == 0) kw0 |= b1;
                else if (slot == 1) kw1 |= b1;
                else if (slot == 2) kw2 |= b1;
                else                kw3 |= b1;
            }
            const unsigned* row = M + (size_t)i * NW;
            #pragma unroll
            for (int s = 0; s < 4; ++s) {
                int w = t + 32 * s;
                if (w < NW) {
                    unsigned add = 0u;
                    if (w > wi) add = row[w];
                    else if (w == wi && bit < 31) add = row[w] & (~0u << (bit + 1));  // only j > i
                    if      (s == 0) rem0 |= add;
                    else if (s == 1) rem1 |= add;
                    else if (s == 2) rem2 |= add;
                    else             rem3 |= add;
                }
            }
        }
    }

    __shared__ unsigned kw[NW];
    __shared__ unsigned basec[NW + 1];
    if (t < NW)      kw[t]      = kw0;
    if (t + 32 < NW) kw[t + 32] = kw1;
    if (t + 64 < NW) kw[t + 64] = kw2;
    if (t + 96 < NW) kw[t + 96] = kw3;
    __syncthreads();
    if (t == 0) {
        unsigned c = 0;
        for (int w = 0; w < NW; ++w) { basec[w] = c; c += __popc(kw[w]); }
        basec[NW] = c;
    }
    __syncthreads();

    float* ob = out_boxes  + (size_t)img * POST_TOP * 4;
    float* os = out_scores + (size_t)img * POST_TOP;
    for (int i = t; i < POST_TOP; i += 32) {
        ob[i * 4 + 0] = 0.f; ob[i * 4 + 1] = 0.f; ob[i * 4 + 2] = 0.f; ob[i * 4 + 3] = 0.f;
        os[i] = -INFINITY;
    }
    __syncthreads();

    const float* bs = box_s   + (size_t)img * K_TOT * 4;
    const float* ss = score_s + (size_t)img * K_TOT;
    for (int s = 0; s < 4; ++s) {
        int w = t + 32 * s;
        if (w >= NW) continue;
        unsigned bits = kw[w];
        unsigned rank = basec[w];
        while (bits) {
            int bp = __ffs(bits) - 1;
            bits &= bits - 1;
            int i = w * 32 + bp;
            if (rank < POST_TOP) {
                ob[rank * 4 + 0] = bs[i * 4 + 0];
                ob[rank * 4 + 1] = bs[i * 4 + 1];
                ob[rank * 4 + 2] = bs[i * 4 + 2];
                ob[rank * 4 + 3] = bs[i * 4 + 3];
                os[rank] = ss[i];
            }
            rank++;
        }
    }
}

// ---------------- host launcher ----------------
static inline size_t alignup256(size_t x) { return (x + 255) & ~(size_t)255; }

extern "C" void kernel_launch(void* const* d_in, const int* in_sizes, int n_in,
                              void* d_out, int out_size, void* d_ws, size_t ws_size,
                              hipStream_t stream) {
    const float* cls0 = (const float*)d_in[0];
    const float* reg0 = (const float*)d_in[1];
    const float* cls1 = (const float*)d_in[2];
    const float* reg1 = (const float*)d_in[3];
    const float* an0  = (const float*)d_in[4];
    const float* an1  = (const float*)d_in[5];
    const int*   ih   = (const int*)d_in[6];
    const int*   iw   = (const int*)d_in[7];

    char* ws = (char*)d_ws;
    size_t off = 0;
    float* scores_all = (float*)(ws + off); off = alignup256(off + sizeof(float) * B_IMGS * N_TOT);
    float* prop_last  = (float*)(ws + off); off = alignup256(off + sizeof(float) * N_TOT * 4);
    int*   top_index  = (int*)  (ws + off); off = alignup256(off + sizeof(int) * B_IMGS * K_TOT);
    float* box_g      = (float*)(ws + off); off = alignup256(off + sizeof(float) * B_IMGS * K_TOT * 4);
    float* score_g    = (float*)(ws + off); off = alignup256(off + sizeof(float) * B_IMGS * K_TOT);
    int*   lvl_g      = (int*)  (ws + off); off = alignup256(off + sizeof(int) * B_IMGS * K_TOT);
    int*   valid_g    = (int*)  (ws + off); off = alignup256(off + sizeof(int) * B_IMGS * K_TOT);
    unsigned* maxc    = (unsigned*)(ws + off); off = alignup256(off + sizeof(unsigned) * B_IMGS);
    float* box_s      = (float*)(ws + off); off = alignup256(off + sizeof(float) * B_IMGS * K_TOT * 4);
    float* score_s    = (float*)(ws + off); off = alignup256(off + sizeof(float) * B_IMGS * K_TOT);
    int*   valid_s    = (int*)  (ws + off); off = alignup256(off + sizeof(int) * B_IMGS * K_TOT);
    float* nb_s       = (float*)(ws + off); off = alignup256(off + sizeof(float) * B_IMGS * K_TOT * 4);
    unsigned* mask    = (unsigned*)(ws + off); off = alignup256(off + sizeof(unsigned) * (size_t)B_IMGS * K_TOT * NW);

    float* out_boxes  = (float*)d_out;
    float* out_scores = out_boxes + (size_t)B_IMGS * POST_TOP * 4;

    k_init<<<(B_IMGS * K_TOT + 255) / 256, 256, 0, stream>>>(maxc, top_index);
    k_decode<<<(N_TOT + 255) / 256, 256, 0, stream>>>(cls0, reg0, cls1, reg1, an0, an1,
                                                      scores_all, prop_last);
    k_topk<<<B_IMGS * 2, 256, 0, stream>>>(scores_all, top_index);
    {
        dim3 g((K_TOT + 255) / 256, B_IMGS);
        k_gather<<<g, 256, 0, stream>>>(top_index, scores_all, prop_last, ih, iw,
                                        box_g, score_g, lvl_g, valid_g, maxc);
    }
    k_sort<<<B_IMGS, 512, 0, stream>>>(box_g, score_g, lvl_g, valid_g, maxc,
                                       box_s, score_s, valid_s, nb_s);
    {
        dim3 g(NW, (K_TOT + 127) / 128, B_IMGS);   // 125 x 32 x 4
        k_mask<<<g, 256, 0, stream>>>(nb_s, mask);
    }
    k_scan<<<B_IMGS, 32, 0, stream>>>(mask, box_s, score_s, valid_s, out_boxes, out_scores);
}